// Model_60550448939680
// MI455X (gfx1250) — compile-verified
//
#include <hip/hip_runtime.h>
#include <math.h>

typedef __attribute__((ext_vector_type(2))) float v2f;
typedef __attribute__((ext_vector_type(8))) float v8f;

#define NB 256
#define NS 336
#define NC 321
#define NP 720
#define WIN 12
#define NF 7
#define INSEG 28
#define HID 16
#define OUTSEG 60

#define BT 14            // batches per core block (2 per wave)
#define NWAVES 7
#define NTHREADS 224
#define K1PAD 60         // 56 (re+im of 28 windows) + 1 bias col + pad
#define N1 32            // 16 re + 16 im hidden
#define K2PAD 36         // 32 + 1 bias col + pad
#define N2COLS 120       // interleaved (re,im) for 60 output segments
#define HSTR 36
#define YSTR 20

__device__ inline v8f wmma_f32(v2f a, v2f b, v8f c) {
  // D = A(16x4 f32) x B(4x16 f32) + C(16x16 f32)
  return __builtin_amdgcn_wmma_f32_16x16x4_f32(false, a, false, b, (short)0, c,
                                               false, false);
}

// ---------------- kernel 0: per-(b,c) mean over S -----------------
__global__ __launch_bounds__(256)
void k_mean(const float* __restrict__ x, float* __restrict__ mean) {
  int b = blockIdx.x;
  int c = blockIdx.y * 256 + threadIdx.x;
  if (c >= NC) return;
  const float* p = x + (size_t)b * NS * NC + c;
  float s = 0.f;
  for (int i = 0; i < NS; ++i) s += p[(size_t)i * NC];
  mean[(size_t)b * NC + c] = s * (1.0f / (float)NS);
}

// ---------- kernel 1: xt[b][c][s] = x[b][s][c] - mean (coalesced) ----------
__global__ __launch_bounds__(256)
void k_transpose_in(const float* __restrict__ x, const float* __restrict__ mean,
                    float* __restrict__ xt) {
  __shared__ float tile[32][33];
  int b = blockIdx.z;
  int s0 = blockIdx.x * 32, c0 = blockIdx.y * 32;
  int tx = threadIdx.x, ty = threadIdx.y;  // 32 x 8
#pragma unroll
  for (int i = 0; i < 4; ++i) {
    int s = s0 + ty + i * 8, c = c0 + tx;
    if (s < NS && c < NC)
      tile[ty + i * 8][tx] = x[((size_t)b * NS + s) * NC + c] - mean[(size_t)b * NC + c];
  }
  __syncthreads();
#pragma unroll
  for (int i = 0; i < 4; ++i) {
    int c = c0 + ty + i * 8, s = s0 + tx;
    if (s < NS && c < NC)
      xt[((size_t)b * NC + c) * NS + s] = tile[tx][ty + i * 8];
  }
}

// ---------------- kernel 2: fused DFT -> GEMM1 -> GEMM2 -> IDFT ----------------
__global__ __launch_bounds__(NTHREADS)
void k_core(const float* __restrict__ xt, const float* __restrict__ mean,
            const float* __restrict__ W1, const float* __restrict__ b1,
            const float* __restrict__ W2, const float* __restrict__ b2,
            float* __restrict__ outT) {
  __shared__ float V1[K1PAD * N1];       // 7680 B
  __shared__ float V2[K2PAD * N2COLS];   // 17280 B
  __shared__ float Dsh[2 * NF * WIN];    // rfft coeffs (re, im)
  __shared__ float Ash[2 * WIN * NF];    // irfft coeffs (re, im)
  __shared__ float pool[BT * NS];        // xs during GEMM1, ybw after sync
  __shared__ float hb[NWAVES * 16 * HSTR];

  const int tid = threadIdx.x;
  const int wave = tid >> 5;
  const int lane = tid & 31;
  const int m = lane & 15;   // row of A / col of B,C,D
  const int kh = lane >> 4;  // K half selector
  const int c = blockIdx.x;
  const int b0 = blockIdx.y * BT;

  // --- DFT / IDFT coefficient tables (ortho norm) ---
  const float inv_s12 = 0.28867513459481287f;  // 1/sqrt(12)
  for (int t = tid; t < 2 * NF * WIN; t += NTHREADS) {
    int sel = t / (NF * WIN);
    int rem = t % (NF * WIN);
    int f = rem / WIN, w = rem % WIN;
    float ang = 0.5235987755982988f * (float)(f * w);  // 2*pi/12
    Dsh[t] = sel ? (-sinf(ang) * inv_s12) : (cosf(ang) * inv_s12);
  }
  for (int t = tid; t < 2 * WIN * NF; t += NTHREADS) {
    int sel = t / (WIN * NF);
    int rem = t % (WIN * NF);
    int w = rem / NF, f = rem % NF;
    float cf = (f == 0 || f == NF - 1) ? 1.0f : 2.0f;
    float ang = 0.5235987755982988f * (float)(f * w);
    Ash[t] = sel ? (-cf * sinf(ang) * inv_s12) : (cf * cosf(ang) * inv_s12);
  }
  // --- zero weight staging (padding rows MUST be 0: 0*garbage can be NaN) ---
  for (int t = tid; t < K1PAD * N1; t += NTHREADS) V1[t] = 0.f;
  for (int t = tid; t < K2PAD * N2COLS; t += NTHREADS) V2[t] = 0.f;
  __syncthreads();

  // --- realify W1/b1 into V1 [60 x 32] ---
  for (int t = tid; t < HID * INSEG; t += NTHREADS) {
    int h = t / INSEG, k = t % INSEG;
    const float* wp = W1 + ((size_t)(c * HID + h) * INSEG + k) * 2;
    float re = wp[0], im = wp[1];
    V1[k * N1 + h] = re;
    V1[(INSEG + k) * N1 + h] = -im;
    V1[k * N1 + HID + h] = im;
    V1[(INSEG + k) * N1 + HID + h] = re;
  }
  if (tid < HID) {
    const float* bp = b1 + ((size_t)c * HID + tid) * 2;
    V1[56 * N1 + tid] = bp[0];
    V1[56 * N1 + HID + tid] = bp[1];
  }
  // --- realify W2/b2 into V2 [36 x 120], columns (re_p, im_p) interleaved ---
  for (int t = tid; t < OUTSEG * HID; t += NTHREADS) {
    int p = t / HID, h = t % HID;
    const float* wp = W2 + ((size_t)(c * OUTSEG + p) * HID + h) * 2;
    float re = wp[0], im = wp[1];
    V2[h * N2COLS + 2 * p] = re;
    V2[(HID + h) * N2COLS + 2 * p] = -im;
    V2[h * N2COLS + 2 * p + 1] = im;
    V2[(HID + h) * N2COLS + 2 * p + 1] = re;
  }
  if (tid < OUTSEG) {
    const float* bp = b2 + ((size_t)c * OUTSEG + tid) * 2;
    V2[32 * N2COLS + 2 * tid] = bp[0];
    V2[32 * N2COLS + 2 * tid + 1] = bp[1];
  }
  // --- stage mean-subtracted series for this block's batches ---
  for (int t = tid; t < BT * NS; t += NTHREADS) {
    int bb = t / NS, s = t % NS;
    int bglob = b0 + bb;
    pool[bb * NS + s] = (bglob < NB) ? xt[((size_t)bglob * NC + c) * NS + s] : 0.f;
  }
  __syncthreads();

  // Row mapping: wave owns 16 rows = 2 batches x 7 freqs + 2 pad rows.
  int mm = (m < 14) ? m : (m - 14);
  const int bloc = 2 * wave + mm / NF;  // local batch 0..13
  const int f = mm % NF;
  const float* xrow = &pool[bloc * NS];
  const float* dR = &Dsh[f * WIN];
  const float* dI = &Dsh[NF * WIN + f * WIN];

  // U[row][kc]: kc<28 -> Re rfft of window kc; kc<56 -> Im; kc==56 -> 1 (bias)
  auto uval = [&](int kc) -> float {
    if (kc == 56) return 1.0f;
    if (kc > 56) return 0.0f;
    const float* dd = (kc < INSEG) ? dR : dI;
    int wi = (kc < INSEG) ? kc : (kc - INSEG);
    const float* xr = xrow + wi * WIN;
    float u = 0.f;
#pragma unroll
    for (int w = 0; w < WIN; ++w) u = fmaf(dd[w], xr[w], u);
    return u;
  };

  // ---- GEMM1: U[16x60] * V1[60x32] -> H[16x32], DFT fused into A-frags ----
  v8f acc1[2] = {{0.f, 0.f, 0.f, 0.f, 0.f, 0.f, 0.f, 0.f},
                 {0.f, 0.f, 0.f, 0.f, 0.f, 0.f, 0.f, 0.f}};
  for (int kk = 0; kk < K1PAD / 4; ++kk) {
    int k0 = kk * 4 + kh * 2;
    v2f a;
    a.x = uval(k0);
    a.y = uval(k0 + 1);
#pragma unroll
    for (int nt = 0; nt < 2; ++nt) {
      v2f bf;
      bf.x = V1[k0 * N1 + nt * HID + m];
      bf.y = V1[(k0 + 1) * N1 + nt * HID + m];
      acc1[nt] = wmma_f32(a, bf, acc1[nt]);
    }
  }
  // spill H to LDS (C-frag layout -> row major), append constant-1 bias column
#pragma unroll
  for (int nt = 0; nt < 2; ++nt)
#pragma unroll
    for (int i = 0; i < 8; ++i)
      hb[(wave * 16 + i + 8 * kh) * HSTR + nt * HID + m] = acc1[nt][i];
  if (lane < 16) {
    float* hr = &hb[(wave * 16 + lane) * HSTR];
    hr[32] = 1.0f;
    hr[33] = 0.f;
    hr[34] = 0.f;
    hr[35] = 0.f;
  }
  __syncthreads();  // all waves done with xs; pool is reused as ybw below

  // ---- GEMM2 per 16-col tile + fused irfft + coalesced store ----
  float* ybw = pool;
  const float* hrow = &hb[(wave * 16 + m) * HSTR];
  const float* ARt = &Ash[0];
  const float* AIt = &Ash[WIN * NF];
  for (int nt = 0; nt < 8; ++nt) {
    v8f acc = {0.f, 0.f, 0.f, 0.f, 0.f, 0.f, 0.f, 0.f};
#pragma unroll
    for (int kk = 0; kk < K2PAD / 4; ++kk) {
      int k0 = kk * 4 + kh * 2;
      v2f a;
      a.x = hrow[k0];
      a.y = hrow[k0 + 1];
      v2f bf;
      bf.x = V2[k0 * N2COLS + nt * HID + m];
      bf.y = V2[(k0 + 1) * N2COLS + nt * HID + m];
      acc = wmma_f32(a, bf, acc);
    }
    // stage Y tile (16 rows x 16 cols = 8 (re,im) segment pairs) per wave
#pragma unroll
    for (int i = 0; i < 8; ++i)
      ybw[(wave * 16 + i + 8 * kh) * YSTR + m] = acc[i];
    // irfft: t[w] = sum_f AR[w][f]*ReY + AI[w][f]*ImY ; add mean; store
    for (int idx = lane; idx < 2 * 8 * WIN; idx += 32) {
      int w = idx % WIN;
      int pp = (idx / WIN) % 8;
      int bb = idx / (8 * WIN);  // 0 or 1 (batch within wave)
      int p = nt * 8 + pp;
      int bglob = b0 + 2 * wave + bb;
      if (p < OUTSEG && bglob < NB) {
        const float* yr0 = &ybw[(wave * 16 + bb * NF) * YSTR];
        float t = 0.f;
#pragma unroll
        for (int ff = 0; ff < NF; ++ff) {
          t = fmaf(ARt[w * NF + ff], yr0[ff * YSTR + 2 * pp], t);
          t = fmaf(AIt[w * NF + ff], yr0[ff * YSTR + 2 * pp + 1], t);
        }
        outT[((size_t)bglob * NC + c) * NP + p * WIN + w] =
            t + mean[(size_t)bglob * NC + c];
      }
    }
  }
}

// ---------------- kernel 3: out[b][t][c] = outT[b][c][t] ----------------
__global__ __launch_bounds__(256)
void k_transpose_out(const float* __restrict__ outT, float* __restrict__ out) {
  __shared__ float tile[32][33];
  int b = blockIdx.z;
  int t0 = blockIdx.x * 32, c0 = blockIdx.y * 32;
  int tx = threadIdx.x, ty = threadIdx.y;
#pragma unroll
  for (int i = 0; i < 4; ++i) {
    int cc = c0 + ty + i * 8, t = t0 + tx;
    if (t < NP && cc < NC)
      tile[ty + i * 8][tx] = outT[((size_t)b * NC + cc) * NP + t];
  }
  __syncthreads();
#pragma unroll
  for (int i = 0; i < 4; ++i) {
    int t = t0 + ty + i * 8, cc = c0 + tx;
    if (t < NP && cc < NC)
      out[((size_t)b * NP + t) * NC + cc] = tile[tx][ty + i * 8];
  }
}

extern "C" void kernel_launch(void* const* d_in, const int* in_sizes, int n_in,
                              void* d_out, int out_size, void* d_ws, size_t ws_size,
                              hipStream_t stream) {
  (void)in_sizes; (void)n_in; (void)out_size; (void)ws_size;
  const float* x  = (const float*)d_in[0];
  const float* W1 = (const float*)d_in[1];  // complex64 as float pairs
  const float* b1 = (const float*)d_in[2];
  const float* W2 = (const float*)d_in[3];
  const float* b2 = (const float*)d_in[4];
  float* out = (float*)d_out;

  float* ws = (float*)d_ws;
  float* mean = ws;                                  // NB*NC
  float* xt = mean + (size_t)NB * NC;                // NB*NC*NS
  float* outT = xt + (size_t)NB * NC * NS;           // NB*NC*NP

  k_mean<<<dim3(NB, (NC + 255) / 256), 256, 0, stream>>>(x, mean);
  k_transpose_in<<<dim3((NS + 31) / 32, (NC + 31) / 32, NB), dim3(32, 8), 0,
                   stream>>>(x, mean, xt);
  k_core<<<dim3(NC, (NB + BT - 1) / BT), NTHREADS, 0, stream>>>(
      xt, mean, W1, b1, W2, b2, outT);
  k_transpose_out<<<dim3((NP + 31) / 32, (NC + 31) / 32, NB), dim3(32, 8), 0,
                    stream>>>(outT, out);
}